// ST_decomposition0_5832565587987
// MI455X (gfx1250) — compile-verified
//
#include <hip/hip_runtime.h>
#include <hip/hip_bf16.h>
#include <math.h>

// Problem constants (from reference setup_inputs)
#define BB   32
#define LL   96
#define NN   256
#define CC   64
#define KK   256
#define ROWS (BB*NN)      // 8192
#define CL   (CC*LL)      // 6144

typedef __attribute__((ext_vector_type(2))) float v2f;
typedef __attribute__((ext_vector_type(8))) float v8f;

// ---------------------------------------------------------------------------
// Kernel 0: fold centers against W/b.
//   colsumW[k][l] = sum_c W[c]*centers[k, c*L+l]      (256 x 96)
//   bdotc[k]     = sum_{c,l} b[c]*centers[k, c*L+l]
//   c2[k]        = ||centers[k]||^2
// One block per k, 96 threads (one per l).
// ---------------------------------------------------------------------------
__global__ __launch_bounds__(96)
void k0_fold_centers(const float* __restrict__ centers,
                     const float* __restrict__ W, const float* __restrict__ bp,
                     float* __restrict__ colsumW, float* __restrict__ bdotc,
                     float* __restrict__ c2)
{
    __shared__ float pb[LL], pc[LL];
    int k = blockIdx.x;
    int l = threadIdx.x;
    const float* cr = centers + (size_t)k * CL;
    float sW = 0.f, sB = 0.f, sC = 0.f;
    for (int c = 0; c < CC; ++c) {
        float v = cr[c*LL + l];
        sW = fmaf(W[c],  v, sW);
        sB = fmaf(bp[c], v, sB);
        sC = fmaf(v, v, sC);
    }
    colsumW[k*LL + l] = sW;
    pb[l] = sB; pc[l] = sC;
    __syncthreads();
    if (l == 0) {
        float b = 0.f, cs = 0.f;
        for (int i = 0; i < LL; ++i) { b += pb[i]; cs += pc[i]; }
        bdotc[k] = b; c2[k] = cs;
    }
}

// ---------------------------------------------------------------------------
// Kernel 1: per-row norm via closed form.
//   ||feats_row||^2 = (sumW2)*sxx + 2*(sumWb)*sx + L*(sumb2)
// ---------------------------------------------------------------------------
__global__ __launch_bounds__(256)
void k1_row_stats(const float* __restrict__ x,
                  const float* __restrict__ W, const float* __restrict__ bp,
                  float* __restrict__ inv_norm, float* __restrict__ z2v)
{
    int row = blockIdx.x * blockDim.x + threadIdx.x;
    if (row >= ROWS) return;
    int bidx = row >> 8, n = row & (NN - 1);
    const float* xp = x + (size_t)bidx * LL * NN + n;
    float sx = 0.f, sxx = 0.f;
    for (int l = 0; l < LL; ++l) {
        float v = xp[l*NN];
        sx += v; sxx = fmaf(v, v, sxx);
    }
    float A = 0.f, Bw = 0.f, Cb = 0.f;
    for (int c = 0; c < CC; ++c) {
        float w = W[c], bv = bp[c];
        A  = fmaf(w, w, A);
        Bw = fmaf(w, bv, Bw);
        Cb = fmaf(bv, bv, Cb);
    }
    float n2   = A * sxx + 2.0f * Bw * sx + (float)LL * Cb;
    float norm = sqrtf(n2);
    float inv  = 1.0f / fmaxf(norm, 1e-8f);
    inv_norm[row] = inv;
    float zn = norm * inv;           // == 1 unless norm < eps
    z2v[row] = zn * zn;
}

// ---------------------------------------------------------------------------
// Kernel 2: WMMA GEMM (8192x256x96 f32) + d2 + q + argmin.
// 512 blocks x 512 threads (16 wave32). Block = 16-row stripe.
// Wave w computes the 16x16 tile of centers [16w, 16w+16) with 24
// v_wmma_f32_16x16x4_f32 steps over Kdim=96.
// Fragment layout (ISA 7.12.2, 32-bit):
//   A 16x4 : lane%16 = M, h=lane/16 selects K pair {2h,2h+1}; v0=first,v1=second
//   B 4x16 : lane%16 = N, same K-pair split
//   C/D    : v8f, element r -> (M = r + 8h, N = lane%16)
// ---------------------------------------------------------------------------
__global__ __launch_bounds__(512)
void k2_assign(const float* __restrict__ x,
               const float* __restrict__ colsumW,
               const float* __restrict__ bdotc, const float* __restrict__ c2,
               const float* __restrict__ inv_norm, const float* __restrict__ z2v,
               float* __restrict__ d2_out, float* __restrict__ q_out,
               int* __restrict__ ids)
{
    __shared__ float xs[16 * LL];      // A tile [m][l]      : 6 KB
    __shared__ float ds[16 * KK];      // d2 tile [m][k]     : 16 KB
    __shared__ float qsum[16];
    int tid = threadIdx.x;
    int rb  = blockIdx.x * 16;         // row base (same batch for all 16 rows)
    int bidx = rb >> 8;
    int nb   = rb & (NN - 1);

    // Stage A tile: xs[m][l] = x[bidx, l, nb+m]; consecutive tid -> consecutive m
    for (int i = tid; i < 16 * LL; i += 512) {
        int m = i & 15, l = i >> 4;
        xs[m * LL + l] = x[((size_t)bidx * LL + l) * NN + nb + m];
    }
    __syncthreads();

    int wave = tid >> 5;
    int lane = tid & 31;
    int mlo  = lane & 15;
    int h    = lane >> 4;
    int kc   = wave * 16 + mlo;         // this lane's center column (N index)
    const float* Brow = colsumW + kc * LL;

    v8f acc = {};
    for (int kb = 0; kb < LL / 4; ++kb) {
        int kd = kb * 4 + 2 * h;
        v2f a, b;
        a.x = xs[mlo * LL + kd];
        a.y = xs[mlo * LL + kd + 1];
        b.x = Brow[kd];
        b.y = Brow[kd + 1];
        acc = __builtin_amdgcn_wmma_f32_16x16x4_f32(
            /*neg_a=*/false, a, /*neg_b=*/false, b,
            /*c_mod=*/(short)0, acc, /*reuse_a=*/false, /*reuse_b=*/false);
    }

    // d2 tile into LDS
    float bk = bdotc[kc], ck = c2[kc];
    for (int r = 0; r < 8; ++r) {
        int m   = r + 8 * h;
        int row = rb + m;
        float zc = (acc[r] + bk) * inv_norm[row];   // Z_row . center_k
        float d2 = z2v[row] + ck - 2.0f * zc;
        ds[m * KK + kc] = d2 > 0.0f ? d2 : 0.0f;
    }
    __syncthreads();

    // Per-row argmin (first min, matching jnp.argmin) + q row-sum
    if (tid < 16) {
        int m = tid;
        const float* dr = ds + m * KK;
        float best = dr[0];
        int   bi   = 0;
        float s    = 1.0f / (1.0f + best);
        for (int k = 1; k < KK; ++k) {
            float v = dr[k];
            s += 1.0f / (1.0f + v);
            if (v < best) { best = v; bi = k; }
        }
        qsum[m] = s;
        ids[rb + m] = bi;
    }
    __syncthreads();

    // Write d2 and normalized q, coalesced
    for (int i = tid; i < 16 * KK; i += 512) {
        int m = i >> 8, k = i & (KK - 1);
        float v = ds[m * KK + k];
        int row = rb + m;
        d2_out[(size_t)row * KK + k] = v;
        q_out [(size_t)row * KK + k] = (1.0f / (1.0f + v)) / qsum[m];
    }
}

// ---------------------------------------------------------------------------
// Kernel 3: bandwidth kernel — Z and res (the 402 MB of writes).
//   Z[row, j] = (x[b, j%L, n]*W[j/L] + b[j/L]) * inv_norm[row]
//   res[b,l,n,c] uses column j2 = l*C + c reinterpreted as (c'=j2/L, l'=j2%L)
//   minus centers[ids[row], j2] (centers are L2-resident).
// One block per row, 256 threads.
// ---------------------------------------------------------------------------
__global__ __launch_bounds__(256)
void k3_outputs(const float* __restrict__ x,
                const float* __restrict__ W, const float* __restrict__ bp,
                const float* __restrict__ centers,
                const float* __restrict__ inv_norm, const int* __restrict__ ids,
                float* __restrict__ res_out, float* __restrict__ z_out)
{
    __shared__ float xl[LL], Ws[CC], bs[CC];
    __shared__ float inv_s;
    __shared__ int   id_s;
    int row  = blockIdx.x;
    int bidx = row >> 8, n = row & (NN - 1);
    int tid  = threadIdx.x;

    if (tid < LL)                    xl[tid]        = x[((size_t)bidx * LL + tid) * NN + n];
    if (tid >= 128 && tid < 192)     Ws[tid - 128]  = W[tid - 128];
    if (tid >= 192 && tid < 256)     bs[tid - 192]  = bp[tid - 192];
    if (tid == 96)                   inv_s          = inv_norm[row];
    if (tid == 97)                   id_s           = ids[row];
    __syncthreads();

    float inv = inv_s;
    const float* crow = centers + (size_t)id_s * CL;
    float* zr = z_out + (size_t)row * CL;

    // Z: column j = c*L + l (fully coalesced)
    for (int j = tid; j < CL; j += 256) {
        int c = j / LL, l = j - c * LL;
        zr[j] = (xl[l] * Ws[c] + bs[c]) * inv;
    }
    // res: iterate j2 = l*C + c; value comes from column index j2 itself
    for (int j2 = tid; j2 < CL; j2 += 256) {
        int l  = j2 >> 6, c = j2 & (CC - 1);
        int cp = j2 / LL, lp = j2 - cp * LL;
        float v = (xl[lp] * Ws[cp] + bs[cp]) * inv - crow[j2];
        res_out[(((size_t)bidx * LL + l) * NN + n) * CC + c] = v;
    }
}

// ---------------------------------------------------------------------------
// Launch
// ---------------------------------------------------------------------------
extern "C" void kernel_launch(void* const* d_in, const int* in_sizes, int n_in,
                              void* d_out, int out_size, void* d_ws, size_t ws_size,
                              hipStream_t stream)
{
    const float* x       = (const float*)d_in[0];   // [32,96,256]
    const float* W_proj  = (const float*)d_in[1];   // [64]
    const float* b_proj  = (const float*)d_in[2];   // [64]
    const float* centers = (const float*)d_in[3];   // [256, 6144]

    float* out = (float*)d_out;
    float* res_out = out;                                   // 50,331,648
    float* d2_out  = out + (size_t)BB*LL*NN*CC;             // 2,097,152
    float* z_out   = d2_out + (size_t)ROWS*KK;              // 50,331,648
    float* q_out   = z_out  + (size_t)ROWS*CL;              // 2,097,152

    // Workspace layout (floats): ~199 KB total
    float* ws       = (float*)d_ws;
    float* colsumW  = ws;                       // 256*96 = 24576
    float* bdotc    = colsumW + KK*LL;          // 256
    float* c2       = bdotc + KK;               // 256
    float* inv_norm = c2 + KK;                  // 8192
    float* z2v      = inv_norm + ROWS;          // 8192
    int*   ids      = (int*)(z2v + ROWS);       // 8192

    k0_fold_centers<<<KK, LL, 0, stream>>>(centers, W_proj, b_proj,
                                           colsumW, bdotc, c2);
    k1_row_stats<<<ROWS/256, 256, 0, stream>>>(x, W_proj, b_proj, inv_norm, z2v);
    k2_assign<<<ROWS/16, 512, 0, stream>>>(x, colsumW, bdotc, c2,
                                           inv_norm, z2v, d2_out, q_out, ids);
    k3_outputs<<<ROWS, 256, 0, stream>>>(x, W_proj, b_proj, centers,
                                         inv_norm, ids, res_out, z_out);
}